// FlashMultiHeadAttention_41137196761611
// MI455X (gfx1250) — compile-verified
//
#include <hip/hip_runtime.h>
#include <hip/hip_bf16.h>
#include <math.h>

typedef float v2f __attribute__((ext_vector_type(2)));
typedef float v4f __attribute__((ext_vector_type(4)));
typedef float v8f __attribute__((ext_vector_type(8)));

#define N_HEADS 16
#define HEAD_D  64

static __device__ __forceinline__ v8f wmma_f32(v2f a, v2f b, v8f c) {
    // V_WMMA_F32_16X16X4_F32 : D(16x16 f32) = A(16x4 f32) * B(4x16 f32) + C
    return __builtin_amdgcn_wmma_f32_16x16x4_f32(
        /*neg_a=*/false, a, /*neg_b=*/false, b,
        /*c_mod=*/(short)0, c, /*reuse_a=*/false, /*reuse_b=*/false);
}

// ---------------------------------------------------------------------------
// Generic fp32 WMMA GEMM: C[M,N] = A[M,K] * B[K,N]   (all row-major)
// Block tile: 128(M) x 64(N), 256 threads = 8 waves, each wave -> 16x64.
// K staged through LDS in chunks of 16 with 128-bit loads/stores;
// B tile stored K-major (transposed) so every WMMA fragment is one b64 load.
// ---------------------------------------------------------------------------
__global__ __launch_bounds__(256)
void gemm_f32_wmma(const float* __restrict__ A, const float* __restrict__ B,
                   float* __restrict__ C, int M, int N, int K) {
    __shared__ float As [128][20];  // stride 20: 16B-aligned b128 stores, conflict-free
    __shared__ float Bst[64][18];   // B tile transposed [n][k], even stride

    const int tid  = threadIdx.x;
    const int wave = tid >> 5;
    const int lane = tid & 31;
    const int half = lane >> 4;     // 0: lanes 0-15, 1: lanes 16-31
    const int l16  = lane & 15;

    const int m0 = blockIdx.y * 128;
    const int n0 = blockIdx.x * 64;

    // Per-thread staging coordinates (fixed across K chunks)
    const int rA0 = (0 * 256 + tid) >> 2,  cA0 = ((0 * 256 + tid) & 3) << 2;
    const int rA1 = (1 * 256 + tid) >> 2,  cA1 = ((1 * 256 + tid) & 3) << 2;
    const int rB  = tid >> 4,              cB  = (tid & 15) << 2;

    v8f acc[4] = {};                // 4 N-subtiles of 16 -> 16x64 per wave

    for (int k0 = 0; k0 < K; k0 += 16) {
        // ---- issue all global loads first (b128), then stores ----
        v4f a0 = *(const v4f*)&A[(size_t)(m0 + rA0) * K + (k0 + cA0)];
        v4f a1 = *(const v4f*)&A[(size_t)(m0 + rA1) * K + (k0 + cA1)];
        v4f b0 = *(const v4f*)&B[(size_t)(k0 + rB) * N + (n0 + cB)];

        if (k0 + 16 < K) {          // prefetch next chunk (one per 64B of A, one per 128B of B)
            if (cA0 == 0) __builtin_prefetch(&A[(size_t)(m0 + rA0) * K + (k0 + 16)], 0, 3);
            if (cA1 == 0) __builtin_prefetch(&A[(size_t)(m0 + rA1) * K + (k0 + 16)], 0, 3);
            if ((cB & 31) == 0)
                __builtin_prefetch(&B[(size_t)(k0 + 16 + rB) * N + (n0 + cB)], 0, 3);
        }

        *(v4f*)&As[rA0][cA0] = a0;                    // ds_store_b128
        *(v4f*)&As[rA1][cA1] = a1;                    // ds_store_b128
        Bst[cB + 0][rB] = b0.x;                       // transpose scatter
        Bst[cB + 1][rB] = b0.y;
        Bst[cB + 2][rB] = b0.z;
        Bst[cB + 3][rB] = b0.w;
        __syncthreads();

        const int mrow = wave * 16 + l16;
        #pragma unroll
        for (int kk = 0; kk < 16; kk += 4) {
            // A frag (16x4): lanes 0-15 hold K=kk,kk+1 ; lanes 16-31 K=kk+2,kk+3
            v2f a = *(const v2f*)&As[mrow][kk + half * 2];
            #pragma unroll
            for (int nt = 0; nt < 4; ++nt) {
                // B frag (4x16): lanes 0-15 rows kk,kk+1 ; lanes 16-31 kk+2,kk+3
                v2f b = *(const v2f*)&Bst[nt * 16 + l16][kk + half * 2];
                acc[nt] = wmma_f32(a, b, acc[nt]);
            }
        }
        __syncthreads();
    }

    // C/D layout: lanes 0-15 => N=l16, VGPR v = row v ; lanes 16-31 => row 8+v
    #pragma unroll
    for (int nt = 0; nt < 4; ++nt) {
        #pragma unroll
        for (int v = 0; v < 8; ++v) {
            int m = m0 + wave * 16 + half * 8 + v;
            int n = n0 + nt * 16 + l16;
            C[(size_t)m * N + n] = acc[nt][v];
        }
    }
}

// ---------------------------------------------------------------------------
// RoPE on q,k + split qkv[B,T,3,H,D]:
//   Q,K -> [B,H,T,D]   (row-major per head)
//   V   -> [B,H,D,T]   (TRANSPOSED: K-major so the P*V B-fragment in the
//                       attention kernel is one contiguous 8-byte load)
// One thread per (b,h,t, pair j).
// ---------------------------------------------------------------------------
__global__ __launch_bounds__(256)
void rope_split(const float* __restrict__ qkv,
                float* __restrict__ Q, float* __restrict__ Kd, float* __restrict__ Vt,
                int B, int H, int T, int D) {
    const int halfD = D >> 1;  // 32
    int idx = blockIdx.x * blockDim.x + threadIdx.x;
    int j = idx % halfD;
    int t = (idx / halfD) % T;
    int h = (idx / (halfD * T)) % H;
    int b =  idx / (halfD * T * H);
    if (b >= B) return;

    const int C  = H * D;
    const int bh = b * H + h;
    const float* base = qkv + ((size_t)(b * T + t) * 3) * C + h * D;

    float inv_freq = __powf(10000.0f, -(2.0f * (float)j) / (float)D);
    float ang = (float)t * inv_freq;
    float s, c;
    __sincosf(ang, &s, &c);

    size_t o = ((size_t)bh * T + t) * D + 2 * j;

    float q1 = base[2 * j], q2 = base[2 * j + 1];
    Q[o]     = q1 * c - q2 * s;
    Q[o + 1] = q1 * s + q2 * c;

    float k1 = base[C + 2 * j], k2 = base[C + 2 * j + 1];
    Kd[o]     = k1 * c - k2 * s;
    Kd[o + 1] = k1 * s + k2 * c;

    size_t ov = ((size_t)bh * D + 2 * j) * T + t;   // transposed [B,H,D,T]
    Vt[ov]     = base[2 * C + 2 * j];
    Vt[ov + T] = base[2 * C + 2 * j + 1];
}

// ---------------------------------------------------------------------------
// Flash attention, causal. One wave owns a 16-row Q tile; 8 waves/block.
// grid.x = T/128 q-tiles, grid.y = B*H.
// Scores: S = (Q*scale) x K^T via 16x v_wmma_f32_16x16x4_f32 (D=64)
// Online softmax with half-wave shfl reductions; P re-shaped via per-wave LDS
// into A-fragment layout; O += P x V (V pre-transposed) via 16x WMMA per tile.
// ---------------------------------------------------------------------------
__global__ __launch_bounds__(256)
void flash_attn(const float* __restrict__ Q, const float* __restrict__ K,
                const float* __restrict__ Vt, float* __restrict__ O,
                int B, int H, int T, int D) {
    __shared__ float P_lds[8][16 * 18];   // per-wave 16x16 P tile, even-stride pad

    const int tid  = threadIdx.x;
    const int wave = tid >> 5;
    const int lane = tid & 31;
    const int half = lane >> 4;
    const int l16  = lane & 15;

    const int bh = blockIdx.y;
    const int b  = bh / H;
    const int h  = bh % H;
    const int q0 = blockIdx.x * 128 + wave * 16;
    const int C  = H * D;

    const float* Qp  = Q  + (size_t)bh * T * D;
    const float* Kp  = K  + (size_t)bh * T * D;
    const float* Vtp = Vt + (size_t)bh * D * T;   // [D][T]

    const float scale = rsqrtf((float)D);   // 1/8

    // Preload Q A-fragments (pre-scaled): 16 K-steps of 4
    v2f qa[16];
    #pragma unroll
    for (int s = 0; s < 16; ++s) {
        v2f q = *(const v2f*)(Qp + (size_t)(q0 + l16) * D + 4 * s + half * 2);
        qa[s].x = q.x * scale;
        qa[s].y = q.y * scale;
    }

    v8f o_acc[4] = {};
    float m_i[8], l_i[8];
    #pragma unroll
    for (int v = 0; v < 8; ++v) { m_i[v] = -3.0e38f; l_i[v] = 0.0f; }

    const int kv_end = q0 + 16;             // causal: keys <= last row of tile
    for (int kv0 = 0; kv0 < kv_end; kv0 += 16) {
        // ---- scores S (16x16) ----
        v8f s_acc = {};
        #pragma unroll
        for (int s = 0; s < 16; ++s) {
            // B frag of K^T: lanes 0-15 K[kv0+l16][4s,4s+1], lanes 16-31 +2,+3
            v2f bf = *(const v2f*)(Kp + (size_t)(kv0 + l16) * D + 4 * s + half * 2);
            s_acc = wmma_f32(qa[s], bf, s_acc);
        }

        // ---- causal mask (only on the diagonal tile) ----
        if (kv0 + 15 > q0) {
            #pragma unroll
            for (int v = 0; v < 8; ++v) {
                int m = q0 + half * 8 + v;
                int n = kv0 + l16;
                if (n > m) s_acc[v] = -3.0e38f;
            }
        }

        // ---- online softmax (row stats live per-VGPR, per lane-half) ----
        float alpha[8];
        #pragma unroll
        for (int v = 0; v < 8; ++v) {
            float rmax = s_acc[v];
            #pragma unroll
            for (int off = 1; off < 16; off <<= 1)
                rmax = fmaxf(rmax, __shfl_xor(rmax, off, 32));
            float m_new = fmaxf(m_i[v], rmax);
            alpha[v] = __expf(m_i[v] - m_new);
            m_i[v] = m_new;
            float p = __expf(s_acc[v] - m_new);
            s_acc[v] = p;
            float rsum = p;
            #pragma unroll
            for (int off = 1; off < 16; off <<= 1)
                rsum += __shfl_xor(rsum, off, 32);
            l_i[v] = l_i[v] * alpha[v] + rsum;
        }

        // ---- rescale O, spill P to LDS (D-layout -> A-layout transpose) ----
        #pragma unroll
        for (int v = 0; v < 8; ++v)
            P_lds[wave][(half * 8 + v) * 18 + l16] = s_acc[v];
        #pragma unroll
        for (int nt = 0; nt < 4; ++nt)
            #pragma unroll
            for (int v = 0; v < 8; ++v)
                o_acc[nt][v] *= alpha[v];
        // per-wave LDS ops are in-order; just stop the compiler reordering
        asm volatile("" ::: "memory");

        // ---- O += P x V  (V is [D][T]: fragment = contiguous b64 load) ----
        #pragma unroll
        for (int s4 = 0; s4 < 4; ++s4) {
            int kk = 4 * s4;
            v2f a = *(const v2f*)&P_lds[wave][l16 * 18 + kk + half * 2];
            #pragma unroll
            for (int nt = 0; nt < 4; ++nt) {
                v2f bf = *(const v2f*)(Vtp + (size_t)(nt * 16 + l16) * T
                                           + kv0 + kk + half * 2);
                o_acc[nt] = wmma_f32(a, bf, o_acc[nt]);
            }
        }
        asm volatile("" ::: "memory");
    }

    // ---- normalize and store, fused [B,H,T,D] -> [B,T,C] transpose ----
    float inv_l[8];
    #pragma unroll
    for (int v = 0; v < 8; ++v) inv_l[v] = 1.0f / l_i[v];
    #pragma unroll
    for (int nt = 0; nt < 4; ++nt) {
        #pragma unroll
        for (int v = 0; v < 8; ++v) {
            int t = q0 + half * 8 + v;
            O[((size_t)(b * T + t)) * C + h * D + nt * 16 + l16] = o_acc[nt][v] * inv_l[v];
        }
    }
}

// ---------------------------------------------------------------------------
extern "C" void kernel_launch(void* const* d_in, const int* in_sizes, int n_in,
                              void* d_out, int out_size, void* d_ws, size_t ws_size,
                              hipStream_t stream) {
    const int B = 2, T = 2048, C = 1024, H = N_HEADS, D = HEAD_D;
    const int M  = B * T;          // 4096
    const int N3 = 3 * C;          // 3072

    const float* x     = (const float*)d_in[0];
    const float* w_qkv = (const float*)d_in[1];
    const float* w_out = (const float*)d_in[2];
    float* out = (float*)d_out;

    char* ws = (char*)d_ws;
    float* qkv   = (float*)(ws);                                  // 48 MB: [M, 3C]
    float* Qb    = (float*)(ws + (size_t)48 * 1024 * 1024);       // 16 MB: [B,H,T,D]
    float* Kb    = (float*)(ws + (size_t)64 * 1024 * 1024);       // 16 MB: [B,H,T,D]
    float* Vtb   = (float*)(ws + (size_t)80 * 1024 * 1024);       // 16 MB: [B,H,D,T]
    float* attnO = qkv;                                           // reuse qkv region

    // 1) qkv = x @ w_qkv        (4096 x 1024) x (1024 x 3072)
    gemm_f32_wmma<<<dim3(N3 / 64, M / 128), 256, 0, stream>>>(x, w_qkv, qkv, M, N3, C);

    // 2) RoPE(q,k) + split; V stored transposed [B,H,D,T]
    {
        int total = B * H * T * (D / 2);   // 2,097,152
        rope_split<<<total / 256, 256, 0, stream>>>(qkv, Qb, Kb, Vtb, B, H, T, D);
    }

    // 3) causal flash attention -> attnO [B,T,C]
    flash_attn<<<dim3(T / 128, B * H), 256, 0, stream>>>(Qb, Kb, Vtb, attnO, B, H, T, D);

    // 4) out = attnO @ w_out    (4096 x 1024) x (1024 x 1024)
    gemm_f32_wmma<<<dim3(C / 64, M / 128), 256, 0, stream>>>(attnO, w_out, out, M, C, C);
}